// InfoNCE_Loss_10514079941234
// MI455X (gfx1250) — compile-verified
//
#include <hip/hip_runtime.h>
#include <math.h>

typedef float v2f __attribute__((ext_vector_type(2)));
typedef float v8f __attribute__((ext_vector_type(8)));
typedef int vs4i __attribute__((vector_size(16)));        // b128 payload type
typedef __attribute__((address_space(1))) vs4i* gb128_t;  // global
typedef __attribute__((address_space(3))) vs4i* lb128_t;  // LDS

#define DIM 128
#define PANEL_COLS 32
#define LDS_STRIDE 132  // 128 + 4 pad -> lanes hit banks 4 apart: conflict-free
#define PANEL_OFF (PANEL_COLS * LDS_STRIDE)

#if __has_builtin(__builtin_amdgcn_global_load_async_to_lds_b128)
#define HAS_ASYNC_LDS 1
#else
#define HAS_ASYNC_LDS 0
#endif

__device__ __forceinline__ const float* zrow(const float* __restrict__ zi,
                                             const float* __restrict__ zj,
                                             int r, int B) {
  return (r < B) ? (zi + (size_t)r * DIM) : (zj + (size_t)(r - B) * DIM);
}

__device__ __forceinline__ void wait_asynccnt0() {
#if __has_builtin(__builtin_amdgcn_s_wait_asynccnt)
  __builtin_amdgcn_s_wait_asynccnt(0);
#else
  asm volatile("s_wait_asynccnt 0x0" ::: "memory");
#endif
}

// Stage one 32-col x 128-K panel of z into an LDS buffer via async DMA.
__device__ __forceinline__ void stage_panel(const float* __restrict__ zi,
                                            const float* __restrict__ zj,
                                            float* __restrict__ lds_buf,
                                            int cbase, int B, int tid) {
  const int col = tid >> 3;       // 8 threads per column
  const int k0 = (tid & 7) * 16;  // 16 K-values (4x b128) each
  const float* src = zrow(zi, zj, cbase + col, B) + k0;
  float* dst = lds_buf + col * LDS_STRIDE + k0;
#if HAS_ASYNC_LDS
  gb128_t gsrc = (gb128_t)src;
  lb128_t ldst = (lb128_t)dst;
#pragma unroll
  for (int i = 0; i < 4; ++i)
    __builtin_amdgcn_global_load_async_to_lds_b128(gsrc + i, ldst + i, 0, 0);
#else
  float4 tmp[4];
#pragma unroll
  for (int i = 0; i < 4; ++i) tmp[i] = *(const float4*)(src + 4 * i);
#pragma unroll
  for (int i = 0; i < 4; ++i) *(float4*)(dst + 4 * i) = tmp[i];
#endif
}

// Online (max, sum) update of one 16x16 tile result held in C-layout regs.
__device__ __forceinline__ void online_update(const v8f& acc, int colbase,
                                              int row_base, int pos_base,
                                              int l16, int hi,
                                              float* __restrict__ m,
                                              float* __restrict__ ssum,
                                              float* __restrict__ pos) {
  const float SCALE = 14.4269504088896340736f;  // (1/T) * log2(e), T = 0.1
  const bool on_diag = (colbase == row_base);
  const bool on_pos = (colbase == pos_base);
#pragma unroll
  for (int v = 0; v < 8; ++v) {
    const bool hit = (l16 == v + (hi << 3));  // col_in_tile == row_in_tile
    float val = acc[v] * SCALE;               // base-2 logits
    if (on_pos && hit) pos[v] = val;
    if (on_diag && hit) val = -__builtin_inff();  // exclude self-sim
    float nm = fmaxf(m[v], val);
    ssum[v] = ssum[v] * exp2f(m[v] - nm) + exp2f(val - nm);
    m[v] = nm;
  }
}

// One wave = one 16-row tile of sim = (z z^T)/T, fused online softmax.
__global__ __launch_bounds__(256) void infonce_sim_softmax(
    const float* __restrict__ zi, const float* __restrict__ zj,
    float* __restrict__ row_loss, int B) {
  const int n = 2 * B;
  const int tid = threadIdx.x;
  const int lane = tid & 31;
  const int wave = tid >> 5;
  const int hi = lane >> 4;  // half-wave select
  const int l16 = lane & 15;

  const int row_tile = blockIdx.x * 8 + wave;
  const int row_base = row_tile * 16;

  __shared__ float lds[2 * PANEL_OFF];  // double-buffered panels (33.8 KB)

  // ---- Preload A fragments (16 rows x 128 K), f32 WMMA A-layout:
  // frag f: lanes 0-15 hold K=4f+{0,1} of row M=l16; lanes 16-31 hold K=4f+{2,3}.
  const float* arow = zrow(zi, zj, row_base + l16, B);
  v2f a[32];
#pragma unroll
  for (int f = 0; f < 32; ++f) a[f] = *(const v2f*)(arow + 4 * f + 2 * hi);

  float m[8], ssum[8], pos[8];
#pragma unroll
  for (int v = 0; v < 8; ++v) {
    m[v] = -1e30f;  // finite lower bound (avoids inf-inf NaN)
    ssum[v] = 0.f;
    pos[v] = -__builtin_inff();
  }

  const int pos_base = (row_base + B) % n;  // tile base holding the positives
  const int npanels = n / PANEL_COLS;

  // ---- software pipeline: DMA panel p+1 while computing panel p ----
  stage_panel(zi, zj, lds, 0, B, tid);
  wait_asynccnt0();
  __syncthreads();

  for (int p = 0; p < npanels; ++p) {
    float* buf = lds + (p & 1) * PANEL_OFF;
    if (p + 1 < npanels)
      stage_panel(zi, zj, lds + ((p + 1) & 1) * PANEL_OFF,
                  (p + 1) * PANEL_COLS, B, tid);

    // ---- two 16x16 sub-tiles computed as interleaved independent chains,
    // with a one-step B prefetch so LDS latency overlaps the WMMAs ----
    const int colbase0 = p * PANEL_COLS;
    const int colbase1 = colbase0 + 16;
    const float* bcol0 = buf + l16 * LDS_STRIDE + 2 * hi;
    const float* bcol1 = buf + (16 + l16) * LDS_STRIDE + 2 * hi;
    v8f acc0 = {0.f, 0.f, 0.f, 0.f, 0.f, 0.f, 0.f, 0.f};
    v8f acc1 = {0.f, 0.f, 0.f, 0.f, 0.f, 0.f, 0.f, 0.f};
    v2f b0 = *(const v2f*)(bcol0);
    v2f b1 = *(const v2f*)(bcol1);
#pragma unroll
    for (int f = 0; f < 32; ++f) {
      v2f nb0 = b0, nb1 = b1;
      if (f < 31) {  // issue next fragment loads before this step's WMMAs
        nb0 = *(const v2f*)(bcol0 + 4 * (f + 1));
        nb1 = *(const v2f*)(bcol1 + 4 * (f + 1));
      }
      acc0 = __builtin_amdgcn_wmma_f32_16x16x4_f32(false, a[f], false, b0,
                                                   (short)0, acc0, false,
                                                   false);
      acc1 = __builtin_amdgcn_wmma_f32_16x16x4_f32(false, a[f], false, b1,
                                                   (short)0, acc1, false,
                                                   false);
      b0 = nb0;
      b1 = nb1;
    }
    online_update(acc0, colbase0, row_base, pos_base, l16, hi, m, ssum, pos);
    online_update(acc1, colbase1, row_base, pos_base, l16, hi, m, ssum, pos);

    // My panel-(p+1) DMAs must land before anyone reads them next iteration;
    // buffer p is only overwritten after the *next* barrier, so all reads of
    // it are fenced by this one.
    wait_asynccnt0();
    __syncthreads();
  }

  // ---- merge the 16 per-column partials of each row (16-lane shuffle tree) --
#pragma unroll
  for (int v = 0; v < 8; ++v) {
    float mv = m[v], sv = ssum[v], pv = pos[v];
#pragma unroll
    for (int off = 8; off; off >>= 1) {  // masks 1,2,4,8 stay within half-wave
      float om = __shfl_xor(mv, off);
      float os = __shfl_xor(sv, off);
      float op = __shfl_xor(pv, off);
      float nm = fmaxf(mv, om);
      sv = sv * exp2f(mv - nm) + os * exp2f(om - nm);
      mv = nm;
      pv = fmaxf(pv, op);  // exactly one lane saw the positive
    }
    if (l16 == 0) {
      // logits = [pos] ++ negatives(pos included, diag excluded):
      // lse2 = m + log2(sum_neg + 2^(pos-m)); loss = ln2 * (lse2 - pos)
      float lse2 = mv + log2f(sv + exp2f(pv - mv));
      row_loss[row_base + v + (hi << 3)] =
          0.69314718055994530942f * (lse2 - pv);
    }
  }
}

// Deterministic fixed-order mean reduction (single block).
__global__ __launch_bounds__(256) void infonce_reduce(
    const float* __restrict__ row_loss, float* __restrict__ out, int n) {
  __shared__ float sdata[256];
  const int t = threadIdx.x;
  float acc = 0.f;
  for (int i = t; i < n; i += 256) acc += row_loss[i];
  sdata[t] = acc;
  __syncthreads();
  for (int sft = 128; sft; sft >>= 1) {
    if (t < sft) sdata[t] += sdata[t + sft];
    __syncthreads();
  }
  if (t == 0) out[0] = sdata[0] / (float)n;
}

extern "C" void kernel_launch(void* const* d_in, const int* in_sizes, int n_in,
                              void* d_out, int out_size, void* d_ws,
                              size_t ws_size, hipStream_t stream) {
  const float* zi = (const float*)d_in[0];
  const float* zj = (const float*)d_in[1];
  float* out = (float*)d_out;
  float* row_loss = (float*)d_ws;   // n floats of scratch
  const int B = in_sizes[0] / DIM;  // 4096
  const int n = 2 * B;              // 8192
  dim3 block(256);                  // 8 wave32 per block
  dim3 grid(n / 128);               // 8 waves x 16 rows per block
  hipLaunchKernelGGL(infonce_sim_softmax, grid, block, 0, stream, zi, zj,
                     row_loss, B);
  hipLaunchKernelGGL(infonce_reduce, dim3(1), block, 0, stream, row_loss, out,
                     n);
}